// CustomDeepseekV2MoE_78022375899549
// MI455X (gfx1250) — compile-verified
//
#include <hip/hip_runtime.h>
#include <math.h>

// ---- problem config (matches reference) ----
#define TNUM   2048   // tokens
#define HDIM   2048   // hidden
#define ENUM   64     // routed experts
#define IDIM   1024   // moe intermediate
#define ISDIM  2048   // shared intermediate (I * N_SHARED)
#define TOPK   6
#define NGRP   8
#define TOPKG  3
#define RSCALE 2.5f

// ---- GEMM tiling ----
#define TM 32         // token tile
#define TN 128        // out-column tile
#define KT 32         // K chunk
#define ASTR 36       // LDS A row stride (floats), 16B aligned, conflict-free
#define BSTR 136      // LDS B row stride (floats), 16B aligned, conflict-free

typedef float v2f __attribute__((ext_vector_type(2)));
typedef float v8f __attribute__((ext_vector_type(8)));

__device__ __forceinline__ v8f wmma4(v2f a, v2f b, v8f c) {
  // D = A(16x4) * B(4x16) + C, fp32 end-to-end
  return __builtin_amdgcn_wmma_f32_16x16x4_f32(false, a, false, b, (short)0, c,
                                               false, false);
}

__device__ __forceinline__ float silu(float g) {
  return g / (1.0f + __expf(-g));
}

// ---------------------------------------------------------------------------
// init: zero per-expert counts
// ---------------------------------------------------------------------------
__global__ __launch_bounds__(64) void k_init(int* __restrict__ counts) {
  if (threadIdx.x < ENUM) counts[threadIdx.x] = 0;
}

// ---------------------------------------------------------------------------
// router: logits, sigmoid+bias, grouped top-k, scatter to per-expert lists
// one 64-thread block per token
// ---------------------------------------------------------------------------
__global__ __launch_bounds__(64) void k_router(
    const float* __restrict__ x, const float* __restrict__ gw,
    const float* __restrict__ ebias, int* __restrict__ counts,
    int* __restrict__ tok, float* __restrict__ wgt) {
  __shared__ float xs[HDIM];
  __shared__ float sb[ENUM];    // sigmoid scores (for weights)
  __shared__ float scb[ENUM];   // scores + bias (for selection)
  const int t = blockIdx.x;
  const int tid = threadIdx.x;

  for (int i = tid; i < HDIM; i += 64) xs[i] = x[(size_t)t * HDIM + i];
  __syncthreads();

  float acc = 0.f;
  for (int h = 0; h < HDIM; h += 4) {
    acc += xs[h + 0] * gw[(size_t)(h + 0) * ENUM + tid];
    acc += xs[h + 1] * gw[(size_t)(h + 1) * ENUM + tid];
    acc += xs[h + 2] * gw[(size_t)(h + 2) * ENUM + tid];
    acc += xs[h + 3] * gw[(size_t)(h + 3) * ENUM + tid];
  }
  float s = 1.0f / (1.0f + __expf(-acc));
  sb[tid] = s;
  scb[tid] = s + ebias[tid];
  __syncthreads();

  if (tid == 0) {
    // group scores: sum of top-2 (score+bias) within each group of 8
    float gs[NGRP];
    for (int g = 0; g < NGRP; ++g) {
      float m1 = -1e30f, m2 = -1e30f;
      for (int j = 0; j < ENUM / NGRP; ++j) {
        float v = scb[g * (ENUM / NGRP) + j];
        if (v > m1) { m2 = m1; m1 = v; }
        else if (v > m2) { m2 = v; }
      }
      gs[g] = m1 + m2;
    }
    // top-3 groups
    unsigned gmask = 0;
    for (int it = 0; it < TOPKG; ++it) {
      int best = 0; float bv = -1e30f;
      for (int g = 0; g < NGRP; ++g)
        if (!((gmask >> g) & 1u) && gs[g] > bv) { bv = gs[g]; best = g; }
      gmask |= 1u << best;
    }
    // top-6 experts among allowed groups
    unsigned long long used = 0;
    int idxs[TOPK];
    float wsum = 0.f;
    for (int k = 0; k < TOPK; ++k) {
      int best = 0; float bv = -1e30f;
      for (int e = 0; e < ENUM; ++e) {
        if (!((gmask >> (e >> 3)) & 1u)) continue;
        if ((used >> e) & 1ull) continue;
        if (scb[e] > bv) { bv = scb[e]; best = e; }
      }
      used |= 1ull << best;
      idxs[k] = best;
      wsum += sb[best];
    }
    float inv = 1.0f / wsum;
    for (int k = 0; k < TOPK; ++k) {
      int e = idxs[k];
      int pos = atomicAdd(&counts[e], 1);
      tok[(size_t)e * TNUM + pos] = t;
      wgt[(size_t)e * TNUM + pos] = sb[e] * inv;
    }
  }
}

// ---------------------------------------------------------------------------
// scan: exclusive prefix over counts -> compact act-row offsets
// ---------------------------------------------------------------------------
__global__ __launch_bounds__(32) void k_scan(const int* __restrict__ counts,
                                             int* __restrict__ offsets) {
  if (threadIdx.x == 0) {
    int run = 0;
    for (int e = 0; e < ENUM; ++e) { offsets[e] = run; run += counts[e]; }
  }
}

// ---------------------------------------------------------------------------
// shared expert: sact = silu(x @ sw_gate) * (x @ sw_up)   [T, IS]
// ---------------------------------------------------------------------------
__global__ __launch_bounds__(256) void k_shared_gateup(
    const float* __restrict__ x, const float* __restrict__ wgp,
    const float* __restrict__ wup, float* __restrict__ sact) {
  __shared__ float As[TM * ASTR];
  __shared__ float Bg[KT * BSTR];
  __shared__ float Bu[KT * BSTR];
  const int n0 = blockIdx.x * TN;
  const int m0 = blockIdx.y * TM;
  const int tid = threadIdx.x;
  const int wave = tid >> 5, lane = tid & 31;
  const int wm = wave >> 2, wn = wave & 3;
  const int half = lane >> 4, lml = lane & 15;

  const int arow = tid >> 3;
  const int ac4 = (tid & 7) << 2;
  const float* aptr = x + (size_t)(m0 + arow) * HDIM + ac4;

  v8f cg0 = {}, cg1 = {}, cu0 = {}, cu1 = {};
  for (int k0 = 0; k0 < HDIM; k0 += KT) {
    __syncthreads();
    *(float4*)&As[arow * ASTR + ac4] = *(const float4*)(aptr + k0);
#pragma unroll
    for (int r = 0; r < 4; ++r) {
      int flat = r * 256 + tid;
      int brow = flat >> 5;
      int bc4 = (flat & 31) << 2;
      *(float4*)&Bg[brow * BSTR + bc4] =
          *(const float4*)&wgp[(size_t)(k0 + brow) * ISDIM + n0 + bc4];
      *(float4*)&Bu[brow * BSTR + bc4] =
          *(const float4*)&wup[(size_t)(k0 + brow) * ISDIM + n0 + bc4];
    }
    __syncthreads();
    const float* ap = &As[(wm * 16 + lml) * ASTR + 2 * half];
    const int nb = wn * 32 + lml;
#pragma unroll
    for (int kk = 0; kk < KT; kk += 4) {
      v2f a = *(const v2f*)(ap + kk);
      int kr = (kk + 2 * half) * BSTR;
      v2f bg0, bg1, bu0, bu1;
      bg0.x = Bg[kr + nb];      bg0.y = Bg[kr + BSTR + nb];
      bg1.x = Bg[kr + nb + 16]; bg1.y = Bg[kr + BSTR + nb + 16];
      bu0.x = Bu[kr + nb];      bu0.y = Bu[kr + BSTR + nb];
      bu1.x = Bu[kr + nb + 16]; bu1.y = Bu[kr + BSTR + nb + 16];
      cg0 = wmma4(a, bg0, cg0);
      cg1 = wmma4(a, bg1, cg1);
      cu0 = wmma4(a, bu0, cu0);
      cu1 = wmma4(a, bu1, cu1);
    }
  }
  const int col = n0 + wn * 32 + lml;
#pragma unroll
  for (int r = 0; r < 8; ++r) {
    int row = m0 + wm * 16 + half * 8 + r;
    sact[(size_t)row * ISDIM + col]      = silu(cg0[r]) * cu0[r];
    sact[(size_t)row * ISDIM + col + 16] = silu(cg1[r]) * cu1[r];
  }
}

// ---------------------------------------------------------------------------
// shared expert down: out = sact @ sw_down   (initializes d_out)
// ---------------------------------------------------------------------------
__global__ __launch_bounds__(256) void k_shared_down(
    const float* __restrict__ sact, const float* __restrict__ wd,
    float* __restrict__ out) {
  __shared__ float As[TM * ASTR];
  __shared__ float Bs[KT * BSTR];
  const int n0 = blockIdx.x * TN;
  const int m0 = blockIdx.y * TM;
  const int tid = threadIdx.x;
  const int wave = tid >> 5, lane = tid & 31;
  const int wm = wave >> 2, wn = wave & 3;
  const int half = lane >> 4, lml = lane & 15;

  const int arow = tid >> 3;
  const int ac4 = (tid & 7) << 2;
  const float* aptr = sact + (size_t)(m0 + arow) * ISDIM + ac4;

  v8f c0 = {}, c1 = {};
  for (int k0 = 0; k0 < ISDIM; k0 += KT) {
    __syncthreads();
    *(float4*)&As[arow * ASTR + ac4] = *(const float4*)(aptr + k0);
#pragma unroll
    for (int r = 0; r < 4; ++r) {
      int flat = r * 256 + tid;
      int brow = flat >> 5;
      int bc4 = (flat & 31) << 2;
      *(float4*)&Bs[brow * BSTR + bc4] =
          *(const float4*)&wd[(size_t)(k0 + brow) * HDIM + n0 + bc4];
    }
    __syncthreads();
    const float* ap = &As[(wm * 16 + lml) * ASTR + 2 * half];
    const int nb = wn * 32 + lml;
#pragma unroll
    for (int kk = 0; kk < KT; kk += 4) {
      v2f a = *(const v2f*)(ap + kk);
      int kr = (kk + 2 * half) * BSTR;
      v2f b0, b1;
      b0.x = Bs[kr + nb];      b0.y = Bs[kr + BSTR + nb];
      b1.x = Bs[kr + nb + 16]; b1.y = Bs[kr + BSTR + nb + 16];
      c0 = wmma4(a, b0, c0);
      c1 = wmma4(a, b1, c1);
    }
  }
  const int col = n0 + wn * 32 + lml;
#pragma unroll
  for (int r = 0; r < 8; ++r) {
    int row = m0 + wm * 16 + half * 8 + r;
    out[(size_t)row * HDIM + col]      = c0[r];
    out[(size_t)row * HDIM + col + 16] = c1[r];
  }
}

// ---------------------------------------------------------------------------
// routed gate/up: act[compact_row, :] = silu(Xg) * Xu for gathered tokens
// grid: (I/TN, T/TM, E)
// ---------------------------------------------------------------------------
__global__ __launch_bounds__(256) void k_routed_gateup(
    const float* __restrict__ x, const float* __restrict__ wg,
    const float* __restrict__ wu, const int* __restrict__ tok,
    const int* __restrict__ counts, const int* __restrict__ offsets,
    float* __restrict__ act) {
  const int e = blockIdx.z;
  const int cnt = counts[e];
  const int t0 = blockIdx.y * TM;
  if (t0 >= cnt) return;
  const int aoff = offsets[e];

  __shared__ float As[TM * ASTR];
  __shared__ float Bg[KT * BSTR];
  __shared__ float Bu[KT * BSTR];
  const int n0 = blockIdx.x * TN;
  const int tid = threadIdx.x;
  const int wave = tid >> 5, lane = tid & 31;
  const int wm = wave >> 2, wn = wave & 3;
  const int half = lane >> 4, lml = lane & 15;

  const int arow = tid >> 3;
  const int ac4 = (tid & 7) << 2;
  const int rr = t0 + arow;
  const int tokid = tok[(size_t)e * TNUM + (rr < cnt ? rr : 0)];
  const float* aptr = x + (size_t)tokid * HDIM + ac4;
  const float* wgp = wg + (size_t)e * HDIM * IDIM;
  const float* wup = wu + (size_t)e * HDIM * IDIM;

  v8f cg0 = {}, cg1 = {}, cu0 = {}, cu1 = {};
  for (int k0 = 0; k0 < HDIM; k0 += KT) {
    __syncthreads();
    *(float4*)&As[arow * ASTR + ac4] = *(const float4*)(aptr + k0);
#pragma unroll
    for (int r = 0; r < 4; ++r) {
      int flat = r * 256 + tid;
      int brow = flat >> 5;
      int bc4 = (flat & 31) << 2;
      *(float4*)&Bg[brow * BSTR + bc4] =
          *(const float4*)&wgp[(size_t)(k0 + brow) * IDIM + n0 + bc4];
      *(float4*)&Bu[brow * BSTR + bc4] =
          *(const float4*)&wup[(size_t)(k0 + brow) * IDIM + n0 + bc4];
    }
    __syncthreads();
    const float* ap = &As[(wm * 16 + lml) * ASTR + 2 * half];
    const int nb = wn * 32 + lml;
#pragma unroll
    for (int kk = 0; kk < KT; kk += 4) {
      v2f a = *(const v2f*)(ap + kk);
      int kr = (kk + 2 * half) * BSTR;
      v2f bg0, bg1, bu0, bu1;
      bg0.x = Bg[kr + nb];      bg0.y = Bg[kr + BSTR + nb];
      bg1.x = Bg[kr + nb + 16]; bg1.y = Bg[kr + BSTR + nb + 16];
      bu0.x = Bu[kr + nb];      bu0.y = Bu[kr + BSTR + nb];
      bu1.x = Bu[kr + nb + 16]; bu1.y = Bu[kr + BSTR + nb + 16];
      cg0 = wmma4(a, bg0, cg0);
      cg1 = wmma4(a, bg1, cg1);
      cu0 = wmma4(a, bu0, cu0);
      cu1 = wmma4(a, bu1, cu1);
    }
  }
  const int col = n0 + wn * 32 + lml;
#pragma unroll
  for (int r = 0; r < 8; ++r) {
    int rl = t0 + wm * 16 + half * 8 + r;
    if (rl < cnt) {
      act[(size_t)(aoff + rl) * IDIM + col]      = silu(cg0[r]) * cu0[r];
      act[(size_t)(aoff + rl) * IDIM + col + 16] = silu(cg1[r]) * cu1[r];
    }
  }
}

// ---------------------------------------------------------------------------
// routed down: out[token,:] += RSCALE * gate_w * (act_rows @ w_down[e])
// grid: (H/TN, T/TM, E); hardware fp32 atomic scatter
// ---------------------------------------------------------------------------
__global__ __launch_bounds__(256) void k_routed_down(
    const float* __restrict__ act, const float* __restrict__ wdn,
    const int* __restrict__ tok, const float* __restrict__ wgt,
    const int* __restrict__ counts, const int* __restrict__ offsets,
    float* __restrict__ out) {
  const int e = blockIdx.z;
  const int cnt = counts[e];
  const int t0 = blockIdx.y * TM;
  if (t0 >= cnt) return;
  const int aoff = offsets[e];

  __shared__ float As[TM * ASTR];
  __shared__ float Bs[KT * BSTR];
  const int n0 = blockIdx.x * TN;
  const int tid = threadIdx.x;
  const int wave = tid >> 5, lane = tid & 31;
  const int wm = wave >> 2, wn = wave & 3;
  const int half = lane >> 4, lml = lane & 15;

  const int arow = tid >> 3;
  const int ac4 = (tid & 7) << 2;
  const int rr = t0 + arow;
  const int srow = aoff + (rr < cnt ? rr : 0);
  const float* aptr = act + (size_t)srow * IDIM + ac4;
  const float* wd = wdn + (size_t)e * IDIM * HDIM;

  v8f c0 = {}, c1 = {};
  for (int k0 = 0; k0 < IDIM; k0 += KT) {
    __syncthreads();
    *(float4*)&As[arow * ASTR + ac4] = *(const float4*)(aptr + k0);
#pragma unroll
    for (int r = 0; r < 4; ++r) {
      int flat = r * 256 + tid;
      int brow = flat >> 5;
      int bc4 = (flat & 31) << 2;
      *(float4*)&Bs[brow * BSTR + bc4] =
          *(const float4*)&wd[(size_t)(k0 + brow) * HDIM + n0 + bc4];
    }
    __syncthreads();
    const float* ap = &As[(wm * 16 + lml) * ASTR + 2 * half];
    const int nb = wn * 32 + lml;
#pragma unroll
    for (int kk = 0; kk < KT; kk += 4) {
      v2f a = *(const v2f*)(ap + kk);
      int kr = (kk + 2 * half) * BSTR;
      v2f b0, b1;
      b0.x = Bs[kr + nb];      b0.y = Bs[kr + BSTR + nb];
      b1.x = Bs[kr + nb + 16]; b1.y = Bs[kr + BSTR + nb + 16];
      c0 = wmma4(a, b0, c0);
      c1 = wmma4(a, b1, c1);
    }
  }
  const int col = n0 + wn * 32 + lml;
#pragma unroll
  for (int r = 0; r < 8; ++r) {
    int rl = t0 + wm * 16 + half * 8 + r;
    if (rl < cnt) {
      int tokid = tok[(size_t)e * TNUM + rl];
      float sc = RSCALE * wgt[(size_t)e * TNUM + rl];
      unsafeAtomicAdd(&out[(size_t)tokid * HDIM + col], sc * c0[r]);
      unsafeAtomicAdd(&out[(size_t)tokid * HDIM + col + 16], sc * c1[r]);
    }
  }
}

// ---------------------------------------------------------------------------
// launch
// ---------------------------------------------------------------------------
extern "C" void kernel_launch(void* const* d_in, const int* in_sizes, int n_in,
                              void* d_out, int out_size, void* d_ws,
                              size_t ws_size, hipStream_t stream) {
  const float* x      = (const float*)d_in[0];   // [T,H]
  const float* gate_w = (const float*)d_in[1];   // [H,E]
  const float* e_bias = (const float*)d_in[2];   // [E]
  const float* w_gate = (const float*)d_in[3];   // [E,H,I]
  const float* w_up   = (const float*)d_in[4];   // [E,H,I]
  const float* w_down = (const float*)d_in[5];   // [E,I,H]
  const float* sw_g   = (const float*)d_in[6];   // [H,IS]
  const float* sw_u   = (const float*)d_in[7];   // [H,IS]
  const float* sw_d   = (const float*)d_in[8];   // [IS,H]
  float* out = (float*)d_out;                    // [T,H]

  char* wsb = (char*)d_ws;
  int*   counts  = (int*)(wsb);                                  // 64 ints
  int*   offsets = (int*)(wsb + 256);                            // 64 ints
  int*   tokl    = (int*)(wsb + 1024);                           // E*T ints
  float* wgtl    = (float*)(wsb + 1024 + sizeof(int) * (size_t)ENUM * TNUM);
  float* sact    = (float*)(wsb + (size_t)2 * 1024 * 1024);      // T*IS floats
  float* act     = sact + (size_t)TNUM * ISDIM;                  // (T*TOPK)*I

  k_init<<<1, 64, 0, stream>>>(counts);
  k_router<<<TNUM, 64, 0, stream>>>(x, gate_w, e_bias, counts, tokl, wgtl);
  k_scan<<<1, 32, 0, stream>>>(counts, offsets);

  k_shared_gateup<<<dim3(ISDIM / TN, TNUM / TM), 256, 0, stream>>>(
      x, sw_g, sw_u, sact);
  k_shared_down<<<dim3(HDIM / TN, TNUM / TM), 256, 0, stream>>>(
      sact, sw_d, out);

  k_routed_gateup<<<dim3(IDIM / TN, TNUM / TM, ENUM), 256, 0, stream>>>(
      x, w_gate, w_up, tokl, counts, offsets, act);
  k_routed_down<<<dim3(HDIM / TN, TNUM / TM, ENUM), 256, 0, stream>>>(
      act, w_down, tokl, wgtl, counts, offsets, out);
}